// Model_25744033972961
// MI455X (gfx1250) — compile-verified
//
#include <hip/hip_runtime.h>
#include <hip/hip_bf16.h>
#include <math.h>

typedef float v2f __attribute__((ext_vector_type(2)));
typedef float v8f __attribute__((ext_vector_type(8)));

#define B_DIM 16
#define N_DIM 2048
#define M_DIM 2048

// ---------------------------------------------------------------------------
// Kernel 1: nearest neighbor via V_WMMA_F32_16X16X4_F32.
// One wave per (batch, 16-row src tile). Loops over tgt tiles of 16 cols,
// unrolled 4x: each trip = 12-load clause + 4 independent WMMA/argmax chains.
// argmax of (2*inner - xx - yy) kept per-lane, reduced with 16-lane butterfly.
// ---------------------------------------------------------------------------
__global__ __launch_bounds__(32) void nn_wmma_kernel(
    const float* __restrict__ src, const float* __restrict__ tgt,
    int* __restrict__ idx) {
  const int ntiles = N_DIM / 16;
  const int b  = blockIdx.x / ntiles;
  const int nt = blockIdx.x % ntiles;
  const int lane = threadIdx.x;      // 0..31
  const int half = lane >> 4;        // 0: K0/K1 half, 1: K2/K3 half
  const int l    = lane & 15;

  const float* sx = src + (size_t)b * 3 * N_DIM;
  const float* sy = sx + N_DIM;
  const float* sz = sy + N_DIM;
  const float* tx = tgt + (size_t)b * 3 * M_DIM;
  const float* ty = tx + M_DIM;
  const float* tz = ty + M_DIM;

  // --- A matrix (16x4, K = [x,y,z,0]) ---
  const int row = nt * 16 + l;
  const float sxv = sx[row], syv = sy[row], szv = sz[row];
  v2f a;
  a.x = half ? szv : sxv;            // VGPR0: K0 (x) lower half, K2 (z) upper half
  a.y = half ? 0.0f : syv;           // VGPR1: K1 (y) lower half, K3 (pad) upper half
  const float xx_mine = sxv * sxv + syv * syv + szv * szv;

  // xx for the 8 rows this lane's C registers cover (rows v + 8*half)
  float xxv[8];
#pragma unroll
  for (int v = 0; v < 8; ++v)
    xxv[v] = __shfl(xx_mine, v + 8 * half, 32);

  float best[8];
  int   bidx[8];
#pragma unroll
  for (int v = 0; v < 8; ++v) { best[v] = -3.4e38f; bidx[v] = 0; }

#pragma unroll 4
  for (int m = 0; m < M_DIM; m += 16) {
    const int col = m + l;
    const float txv = tx[col], tyv = ty[col], tzv = tz[col];
    // --- B matrix (4x16) ---
    v2f bm;
    bm.x = half ? tzv : txv;
    bm.y = half ? 0.0f : tyv;
    const float yy = txv * txv + tyv * tyv + tzv * tzv;

    v8f c = {};
    c = __builtin_amdgcn_wmma_f32_16x16x4_f32(
        /*neg_a=*/false, a, /*neg_b=*/false, bm,
        /*c_mod=*/(short)0, c, /*reuse_a=*/false, /*reuse_b=*/false);

#pragma unroll
    for (int v = 0; v < 8; ++v) {
      const float nd = 2.0f * c[v] - xxv[v] - yy;  // neg squared distance
      if (nd > best[v]) { best[v] = nd; bidx[v] = col; }
    }
  }

  // Reduce across the 16 lanes of each half (each half owns rows v+8*half).
#pragma unroll
  for (int off = 8; off >= 1; off >>= 1) {
#pragma unroll
    for (int v = 0; v < 8; ++v) {
      const float ov = __shfl_xor(best[v], off, 16);
      const int   oi = __shfl_xor(bidx[v], off, 16);
      if (ov > best[v] || (ov == best[v] && oi < bidx[v])) {
        best[v] = ov; bidx[v] = oi;
      }
    }
  }
  if (l == 0) {
#pragma unroll
    for (int v = 0; v < 8; ++v) {
      const int r = v + 8 * half;
      idx[(size_t)b * N_DIM + nt * 16 + r] = bidx[v];
    }
  }
}

// ---------------------------------------------------------------------------
// Kernel 2: per-batch reduction of the 22 sums needed for weighted Kabsch:
//   [0..2]  Ss     = sum s
//   [3..5]  Sc     = sum corr
//   [6]     Sw     = sum w
//   [7..9]  Sws    = sum w*s
//   [10..12]Swc    = sum w*corr
//   [13..21]Swsc   = sum w * s_i * c_j
// ---------------------------------------------------------------------------
__global__ __launch_bounds__(256) void reduce_sums_kernel(
    const float* __restrict__ src, const float* __restrict__ tgt,
    const float* __restrict__ w, const int* __restrict__ idx,
    float* __restrict__ sums) {
  const int b = blockIdx.x;
  const float* sx = src + (size_t)b * 3 * N_DIM;
  const float* sy = sx + N_DIM;
  const float* sz = sy + N_DIM;
  const float* tx = tgt + (size_t)b * 3 * M_DIM;
  const float* ty = tx + M_DIM;
  const float* tz = ty + M_DIM;
  const float* wv = w + (size_t)b * N_DIM;
  const int*   ib = idx + (size_t)b * N_DIM;

  float acc[22];
#pragma unroll
  for (int k = 0; k < 22; ++k) acc[k] = 0.0f;

  for (int n = threadIdx.x; n < N_DIM; n += blockDim.x) {
    const float s0 = sx[n], s1 = sy[n], s2 = sz[n];
    const int j = ib[n];
    const float c0 = tx[j], c1 = ty[j], c2 = tz[j];
    const float ww = wv[n];
    acc[0] += s0;  acc[1] += s1;  acc[2] += s2;
    acc[3] += c0;  acc[4] += c1;  acc[5] += c2;
    acc[6] += ww;
    acc[7] += ww * s0;  acc[8] += ww * s1;  acc[9] += ww * s2;
    acc[10] += ww * c0; acc[11] += ww * c1; acc[12] += ww * c2;
    acc[13] += ww * s0 * c0; acc[14] += ww * s0 * c1; acc[15] += ww * s0 * c2;
    acc[16] += ww * s1 * c0; acc[17] += ww * s1 * c1; acc[18] += ww * s1 * c2;
    acc[19] += ww * s2 * c0; acc[20] += ww * s2 * c1; acc[21] += ww * s2 * c2;
  }

  // wave32 shuffle reduction
#pragma unroll
  for (int k = 0; k < 22; ++k) {
    for (int off = 16; off >= 1; off >>= 1)
      acc[k] += __shfl_xor(acc[k], off, 32);
  }

  __shared__ float smem[8 * 22];
  const int wid = threadIdx.x >> 5, ln = threadIdx.x & 31;
  if (ln == 0) {
#pragma unroll
    for (int k = 0; k < 22; ++k) smem[wid * 22 + k] = acc[k];
  }
  __syncthreads();
  if (threadIdx.x == 0) {
#pragma unroll
    for (int k = 0; k < 22; ++k) {
      float t = 0.0f;
      for (int wgi = 0; wgi < 8; ++wgi) t += smem[wgi * 22 + k];
      sums[(size_t)b * 22 + k] = t;
    }
  }
}

// ---------------------------------------------------------------------------
// Kernel 3: per-batch 3x3 Kabsch. One lane per batch.
// Jacobi eigendecomposition of H^T H -> V, sigma; U cols = H v / |Hv|,
// u2 = u0 x u1 (det(U)=+1) which makes R = V U^T automatically equal to the
// reference's reflection-corrected rotation in both det signs.
// ---------------------------------------------------------------------------
__global__ __launch_bounds__(32) void kabsch_kernel(
    const float* __restrict__ sums, float* __restrict__ out) {
  const int b = threadIdx.x;
  if (b >= B_DIM) return;
  const float* s = sums + (size_t)b * 22;

  const float invN = 1.0f / (float)N_DIM;
  float mus[3], muc[3], Sws[3], Swc[3];
#pragma unroll
  for (int i = 0; i < 3; ++i) {
    mus[i] = s[i] * invN;
    muc[i] = s[3 + i] * invN;
    Sws[i] = s[7 + i];
    Swc[i] = s[10 + i];
  }
  const float Sw = s[6];

  float H[9];
#pragma unroll
  for (int i = 0; i < 3; ++i)
#pragma unroll
    for (int j = 0; j < 3; ++j)
      H[i * 3 + j] = s[13 + i * 3 + j] - muc[j] * Sws[i] - mus[i] * Swc[j]
                   + Sw * mus[i] * muc[j];

  // A = H^T H (symmetric PSD)
  float A[9];
#pragma unroll
  for (int i = 0; i < 3; ++i)
#pragma unroll
    for (int j = 0; j < 3; ++j)
      A[i * 3 + j] = H[0 * 3 + i] * H[0 * 3 + j] + H[1 * 3 + i] * H[1 * 3 + j]
                   + H[2 * 3 + i] * H[2 * 3 + j];

  float V[9] = {1.f, 0.f, 0.f, 0.f, 1.f, 0.f, 0.f, 0.f, 1.f};

  auto rot = [&](int p, int q) {
    const float apq = A[p * 3 + q];
    if (fabsf(apq) < 1e-20f) return;
    const float app = A[p * 3 + p], aqq = A[q * 3 + q];
    const float tau = (aqq - app) / (2.0f * apq);
    const float t = (tau >= 0.0f ? 1.0f : -1.0f) /
                    (fabsf(tau) + sqrtf(1.0f + tau * tau));
    const float cth = 1.0f / sqrtf(1.0f + t * t);
    const float sth = t * cth;
#pragma unroll
    for (int k = 0; k < 3; ++k) {
      const float akp = A[k * 3 + p], akq = A[k * 3 + q];
      A[k * 3 + p] = cth * akp - sth * akq;
      A[k * 3 + q] = sth * akp + cth * akq;
    }
#pragma unroll
    for (int k = 0; k < 3; ++k) {
      const float apk = A[p * 3 + k], aqk = A[q * 3 + k];
      A[p * 3 + k] = cth * apk - sth * aqk;
      A[q * 3 + k] = sth * apk + cth * aqk;
    }
#pragma unroll
    for (int k = 0; k < 3; ++k) {
      const float vkp = V[k * 3 + p], vkq = V[k * 3 + q];
      V[k * 3 + p] = cth * vkp - sth * vkq;
      V[k * 3 + q] = sth * vkp + cth * vkq;
    }
  };

  for (int sweep = 0; sweep < 12; ++sweep) {
    rot(0, 1); rot(0, 2); rot(1, 2);
  }

  float d[3] = {A[0], A[4], A[8]};
  int ord[3] = {0, 1, 2};
  if (d[ord[0]] < d[ord[1]]) { int t = ord[0]; ord[0] = ord[1]; ord[1] = t; }
  if (d[ord[0]] < d[ord[2]]) { int t = ord[0]; ord[0] = ord[2]; ord[2] = t; }
  if (d[ord[1]] < d[ord[2]]) { int t = ord[1]; ord[1] = ord[2]; ord[2] = t; }

  float Vs[9];
#pragma unroll
  for (int k = 0; k < 3; ++k)
#pragma unroll
    for (int i = 0; i < 3; ++i)
      Vs[i * 3 + k] = V[i * 3 + ord[k]];

  float U[9];
#pragma unroll
  for (int k = 0; k < 2; ++k) {
    const float vx = Vs[0 * 3 + k], vy = Vs[1 * 3 + k], vz = Vs[2 * 3 + k];
    const float ux = H[0] * vx + H[1] * vy + H[2] * vz;
    const float uy = H[3] * vx + H[4] * vy + H[5] * vz;
    const float uz = H[6] * vx + H[7] * vy + H[8] * vz;
    const float len = sqrtf(ux * ux + uy * uy + uz * uz);
    const float inv = (len > 1e-20f) ? 1.0f / len : 0.0f;
    U[0 * 3 + k] = ux * inv; U[1 * 3 + k] = uy * inv; U[2 * 3 + k] = uz * inv;
  }
  // u2 = u0 x u1 -> det(U) = +1 (absorbs the reference's reflection fix)
  U[0 * 3 + 2] = U[1 * 3 + 0] * U[2 * 3 + 1] - U[2 * 3 + 0] * U[1 * 3 + 1];
  U[1 * 3 + 2] = U[2 * 3 + 0] * U[0 * 3 + 1] - U[0 * 3 + 0] * U[2 * 3 + 1];
  U[2 * 3 + 2] = U[0 * 3 + 0] * U[1 * 3 + 1] - U[1 * 3 + 0] * U[0 * 3 + 1];

  // R = V U^T
  float R[9];
#pragma unroll
  for (int i = 0; i < 3; ++i)
#pragma unroll
    for (int j = 0; j < 3; ++j)
      R[i * 3 + j] = Vs[i * 3 + 0] * U[j * 3 + 0] + Vs[i * 3 + 1] * U[j * 3 + 1]
                   + Vs[i * 3 + 2] * U[j * 3 + 2];

  // t = -R * Sws + Swc
  float tv[3];
#pragma unroll
  for (int i = 0; i < 3; ++i)
    tv[i] = -(R[i * 3 + 0] * Sws[0] + R[i * 3 + 1] * Sws[1] + R[i * 3 + 2] * Sws[2])
            + Swc[i];

  // Output: R [B,3,3] then t [B,3], concatenated flat.
#pragma unroll
  for (int k = 0; k < 9; ++k) out[(size_t)b * 9 + k] = R[k];
#pragma unroll
  for (int i = 0; i < 3; ++i) out[(size_t)B_DIM * 9 + b * 3 + i] = tv[i];
}

extern "C" void kernel_launch(void* const* d_in, const int* in_sizes, int n_in,
                              void* d_out, int out_size, void* d_ws, size_t ws_size,
                              hipStream_t stream) {
  const float* src = (const float*)d_in[0];   // [B,3,N]
  const float* tgt = (const float*)d_in[1];   // [B,3,M]
  const float* w   = (const float*)d_in[2];   // [B,1,N]
  float* out = (float*)d_out;                 // [B*9 + B*3]

  int*   idx  = (int*)d_ws;                                        // B*N ints
  float* sums = (float*)((char*)d_ws + (size_t)B_DIM * N_DIM * sizeof(int));

  nn_wmma_kernel<<<B_DIM * (N_DIM / 16), 32, 0, stream>>>(src, tgt, idx);
  reduce_sums_kernel<<<B_DIM, 256, 0, stream>>>(src, tgt, w, idx, sums);
  kabsch_kernel<<<1, 32, 0, stream>>>(sums, out);
}